// GraphEncoderGNN_75161927680330
// MI455X (gfx1250) — compile-verified
//
#include <hip/hip_runtime.h>
#include <hip/hip_bf16.h>

typedef float v2f __attribute__((ext_vector_type(2)));
typedef float v8f __attribute__((ext_vector_type(8)));

#define NEG_SLOPE 0.2f

static __host__ __device__ inline int cdiv(int a, int b) { return (a + b - 1) / b; }

// ---------------------------------------------------------------------------
// FP32 WMMA GEMM: D[M,Nc] = act(A[M,K] @ B[K,Nc] + bias)
// One wave (32 threads) computes a 16x64 strip: 4 accumulator tiles sharing
// one A fragment per k-step (4x A reuse, 4 v_wmma_f32_16x16x4_f32 per step).
// A-frag: lane L holds A[row0 + L%16][k + 2*(L/16) + {0,1}] in {a.x,a.y}
// B-frag: lane L holds B[k + 2*(L/16) + {0,1}][col + L%16]
// C/D:    VGPR r of lane L = D[row0 + r + 8*(L/16)][col + L%16]
// ---------------------------------------------------------------------------
template <int K>
__global__ __launch_bounds__(32) void wmma_gemm_f32(
    const float* __restrict__ A, const float* __restrict__ B,
    const float* __restrict__ bias, float* __restrict__ D,
    int M, int Nc, int act)
{
    const int lane = threadIdx.x;
    const int half = lane >> 4;   // 0 or 1
    const int lm   = lane & 15;
    const int row0 = blockIdx.x * 16;
    const int col0 = blockIdx.y * 64;

    // Clamp (not branch) row for loads so EXEC stays all-ones for WMMA.
    int arow = row0 + lm; if (arow > M - 1) arow = M - 1;
    const float* __restrict__ Ap = A + (long)arow * K;
    const float* __restrict__ Bp = B + col0 + lm;

    // Pull the next block's A panel toward L2 while we compute this one.
    __builtin_prefetch(A + (long)(row0 + 16 + lm) * K, 0, 1);

    v8f acc[4];
#pragma unroll
    for (int t = 0; t < 4; ++t)
        acc[t] = (v8f){0.f, 0.f, 0.f, 0.f, 0.f, 0.f, 0.f, 0.f};

#pragma unroll 2
    for (int k = 0; k < K; k += 4) {
        const int k0 = k + 2 * half;
        v2f a;
        a.x = Ap[k0];
        a.y = Ap[k0 + 1];
        const float* __restrict__ Br0 = Bp + (long)k0 * Nc;
        const float* __restrict__ Br1 = Bp + (long)(k0 + 1) * Nc;
#pragma unroll
        for (int t = 0; t < 4; ++t) {
            v2f b;
            b.x = Br0[16 * t];
            b.y = Br1[16 * t];
            acc[t] = __builtin_amdgcn_wmma_f32_16x16x4_f32(
                /*neg_a=*/false, a, /*neg_b=*/false, b,
                /*c_mod=*/(short)0, acc[t], /*reuse_a=*/false, /*reuse_b=*/false);
        }
    }

#pragma unroll
    for (int t = 0; t < 4; ++t) {
        const int n = col0 + 16 * t + lm;
        const float bv = bias ? bias[n] : 0.f;
#pragma unroll
        for (int r = 0; r < 8; ++r) {
            const int m = row0 + r + 8 * half;
            float v = acc[t][r] + bv;
            if (act) v = fmaxf(v, 0.f);
            if (m < M) D[(long)m * Nc + n] = v;
        }
    }
}

static void launch_gemm(int K, const float* A, const float* B, const float* bias,
                        float* D, int M, int Nc, int act, hipStream_t stream)
{
    dim3 grid(cdiv(M, 16), Nc / 64);
    switch (K) {
    case 64:  wmma_gemm_f32<64><<<grid, 32, 0, stream>>>(A, B, bias, D, M, Nc, act); break;
    case 128: wmma_gemm_f32<128><<<grid, 32, 0, stream>>>(A, B, bias, D, M, Nc, act); break;
    case 256: wmma_gemm_f32<256><<<grid, 32, 0, stream>>>(A, B, bias, D, M, Nc, act); break;
    }
}

// ---------------------------------------------------------------------------
// Per-node attention logits: as[n] = h[n]·a_src, ad[n] = h[n]·a_dst
// ---------------------------------------------------------------------------
__global__ void node_alpha_kernel(const float* __restrict__ h,
                                  const float* __restrict__ a_src,
                                  const float* __restrict__ a_dst,
                                  float* __restrict__ os, float* __restrict__ od,
                                  int Nn, int C)
{
    const int n = blockIdx.x * blockDim.x + threadIdx.x;
    if (n >= Nn) return;
    const float* __restrict__ hp = h + (long)n * C;
    float s = 0.f, d = 0.f;
    for (int c = 0; c < C; ++c) {
        const float v = hp[c];
        s += v * a_src[c];
        d += v * a_dst[c];
    }
    os[n] = s;
    od[n] = d;
}

// monotonic float<->uint encoding for atomicMax on signed floats
__device__ inline unsigned fenc(float f) {
    unsigned u = __float_as_uint(f);
    return (u & 0x80000000u) ? ~u : (u | 0x80000000u);
}
__device__ inline float fdec(unsigned u) {
    return __uint_as_float((u & 0x80000000u) ? (u ^ 0x80000000u) : ~u);
}

__global__ void init_softmax_kernel(unsigned* __restrict__ emax,
                                    float* __restrict__ denom, int Nn)
{
    const int n = blockIdx.x * blockDim.x + threadIdx.x;
    if (n >= Nn) return;
    emax[n] = 0x007FFFFFu;   // fenc(-inf)
    denom[n] = 0.f;
}

__device__ inline void edge_nodes(const int* __restrict__ ei, int idx, int E,
                                  int& s, int& d)
{
    if (idx < E) { s = ei[idx]; d = ei[E + idx]; }
    else         { s = idx - E; d = idx - E; }   // self loops appended
}

__global__ void edge_max_kernel(const int* __restrict__ ei,
                                const float* __restrict__ as,
                                const float* __restrict__ ad,
                                unsigned* __restrict__ emax, int E, int ET)
{
    const int idx = blockIdx.x * blockDim.x + threadIdx.x;
    if (idx >= ET) return;
    int s, d; edge_nodes(ei, idx, E, s, d);
    float e = as[s] + ad[d];
    e = (e > 0.f) ? e : NEG_SLOPE * e;           // leaky_relu
    atomicMax(&emax[d], fenc(e));
}

__global__ void edge_exp_kernel(const int* __restrict__ ei,
                                const float* __restrict__ as,
                                const float* __restrict__ ad,
                                const unsigned* __restrict__ emax,
                                float* __restrict__ ex,
                                float* __restrict__ denom, int E, int ET)
{
    const int idx = blockIdx.x * blockDim.x + threadIdx.x;
    if (idx >= ET) return;
    int s, d; edge_nodes(ei, idx, E, s, d);
    float e = as[s] + ad[d];
    e = (e > 0.f) ? e : NEG_SLOPE * e;
    const float x = __expf(e - fdec(emax[d]));
    ex[idx] = x;
    atomicAdd(&denom[d], x);
}

// Weighted scatter: agg[dst] += (ex/denom[dst]) * h[src].
// Channel-parallel: each thread handles 4 channels (float4 gather of h[src]),
// so with C=128 one wave32 covers exactly one edge's channels (coalesced).
__global__ void edge_agg_kernel(const int* __restrict__ ei,
                                const float* __restrict__ ex,
                                const float* __restrict__ denom,
                                const float* __restrict__ h,
                                float* __restrict__ agg,
                                int E, int ET, int C)
{
    const int chunks = C >> 2;
    const long tid = (long)blockIdx.x * blockDim.x + threadIdx.x;
    const int idx = (int)(tid / chunks);
    if (idx >= ET) return;
    const int c = (int)(tid % chunks) << 2;
    int s, d; edge_nodes(ei, idx, E, s, d);
    const float w = ex[idx] / denom[d];
    const float4 hv = *reinterpret_cast<const float4*>(h + (long)s * C + c);
    float* __restrict__ ap = agg + (long)d * C + c;
    atomicAdd(ap + 0, w * hv.x);
    atomicAdd(ap + 1, w * hv.y);
    atomicAdd(ap + 2, w * hv.z);
    atomicAdd(ap + 3, w * hv.w);
}

__global__ void bias_relu_kernel(float* __restrict__ a,
                                 const float* __restrict__ b, int total, int C)
{
    const int t = blockIdx.x * blockDim.x + threadIdx.x;
    if (t >= total) return;
    a[t] = fmaxf(a[t] + b[t % C], 0.f);
}

// ---------------------------------------------------------------------------
extern "C" void kernel_launch(void* const* d_in, const int* in_sizes, int n_in,
                              void* d_out, int out_size, void* d_ws, size_t ws_size,
                              hipStream_t stream)
{
    const float* x    = (const float*)d_in[0];
    const int*   ei   = (const int*)d_in[1];
    // d_in[2] = batch (unused by the reference computation)
    const float* W1   = (const float*)d_in[3];
    const float* a1s  = (const float*)d_in[4];
    const float* a1d  = (const float*)d_in[5];
    const float* b1   = (const float*)d_in[6];
    const float* W2   = (const float*)d_in[7];
    const float* a2s  = (const float*)d_in[8];
    const float* a2d  = (const float*)d_in[9];
    const float* b2   = (const float*)d_in[10];
    const float* Wm1  = (const float*)d_in[11];
    const float* bm1  = (const float*)d_in[12];
    const float* Wm2  = (const float*)d_in[13];
    const float* bm2  = (const float*)d_in[14];
    float* out = (float*)d_out;

    const int F_IN = 64, C1 = 128, C2 = 64, NHID = 256, NOUT = 256;
    const int N  = in_sizes[0] / F_IN;       // 50000
    const int E  = in_sizes[1] / 2;          // 800000
    const int ET = E + N;                    // edges + self loops
    (void)n_in; (void)out_size; (void)ws_size;

    // workspace layout (floats)
    float* w = (float*)d_ws;
    long off = 0;
    float*    h1    = w + off; off += (long)N * C1;
    float*    agg1  = w + off; off += (long)N * C1;
    float*    h2    = w + off; off += (long)N * C2;
    float*    agg2  = w + off; off += (long)N * C2;
    float*    as_   = w + off; off += N;
    float*    ad_   = w + off; off += N;
    float*    denom = w + off; off += N;
    float*    ex    = w + off; off += ET;
    unsigned* emax  = (unsigned*)(w + off); off += N;
    float*    hm    = h1;  // MLP hidden [N, NHID] reuses h1+agg1 (N*256 floats)

    const int TB = 256;
    const int gN  = cdiv(N, TB);
    const int gET = cdiv(ET, TB);

    // ---- Layer 1: h1 = x @ W1 --------------------------------------------
    launch_gemm(F_IN, x, W1, nullptr, h1, N, C1, 0, stream);
    node_alpha_kernel<<<gN, TB, 0, stream>>>(h1, a1s, a1d, as_, ad_, N, C1);

    hipMemsetAsync(agg1, 0, (size_t)N * C1 * sizeof(float), stream);
    init_softmax_kernel<<<gN, TB, 0, stream>>>(emax, denom, N);
    edge_max_kernel<<<gET, TB, 0, stream>>>(ei, as_, ad_, emax, E, ET);
    edge_exp_kernel<<<gET, TB, 0, stream>>>(ei, as_, ad_, emax, ex, denom, E, ET);
    edge_agg_kernel<<<cdiv(ET * (C1 / 4), TB), TB, 0, stream>>>(
        ei, ex, denom, h1, agg1, E, ET, C1);
    bias_relu_kernel<<<cdiv(N * C1, TB), TB, 0, stream>>>(agg1, b1, N * C1, C1);

    // ---- Layer 2: h2 = relu_out1 @ W2 ------------------------------------
    launch_gemm(C1, agg1, W2, nullptr, h2, N, C2, 0, stream);
    node_alpha_kernel<<<gN, TB, 0, stream>>>(h2, a2s, a2d, as_, ad_, N, C2);

    hipMemsetAsync(agg2, 0, (size_t)N * C2 * sizeof(float), stream);
    init_softmax_kernel<<<gN, TB, 0, stream>>>(emax, denom, N);
    edge_max_kernel<<<gET, TB, 0, stream>>>(ei, as_, ad_, emax, E, ET);
    edge_exp_kernel<<<gET, TB, 0, stream>>>(ei, as_, ad_, emax, ex, denom, E, ET);
    edge_agg_kernel<<<cdiv(ET * (C2 / 4), TB), TB, 0, stream>>>(
        ei, ex, denom, h2, agg2, E, ET, C2);
    bias_relu_kernel<<<cdiv(N * C2, TB), TB, 0, stream>>>(agg2, b2, N * C2, C2);

    // ---- MLP --------------------------------------------------------------
    launch_gemm(C2, agg2, Wm1, bm1, hm, N, NHID, 1, stream);   // fused bias + relu
    launch_gemm(NHID, hm, Wm2, bm2, out, N, NOUT, 0, stream);  // fused bias
}